// SlidingWindowGraphBaseline_60189671686513
// MI455X (gfx1250) — compile-verified
//
#include <hip/hip_runtime.h>
#include <math.h>

// CDNA5 / gfx1250, wave32. FP32 WMMA (V_WMMA_F32_16X16X4_F32) + async
// global->LDS staging (GLOBAL_LOAD_ASYNC_TO_LDS_B128, ASYNCcnt) with
// LDS double buffering.

typedef __attribute__((ext_vector_type(2))) float v2f;
typedef __attribute__((ext_vector_type(8))) float v8f;

#define DD   768   // feature dim
#define TT   512   // tokens per window
#define NWIN 64    // 4 batches * 16 windows

// Async copy of 16 bytes global -> LDS, tracked by ASYNCcnt (no VGPR staging).
__device__ __forceinline__ void async_cp16(unsigned lds_off, const float* g) {
  asm volatile("global_load_async_to_lds_b128 %0, %1, off"
               :: "v"(lds_off), "v"((unsigned long long)(size_t)g)
               : "memory");
}
__device__ __forceinline__ void async_wait0() {
  asm volatile("s_wait_asynccnt 0x0" ::: "memory");
}

// ---------------------------------------------------------------------------
// Kernel 1: reciprocal row norms  rnorm[row] = 1 / max(||h_row||, 1e-8)
// ---------------------------------------------------------------------------
__global__ __launch_bounds__(256) void rnorm_kernel(const float* __restrict__ emb,
                                                    float* __restrict__ rnorm) {
  const int wave = threadIdx.x >> 5;
  const int lane = threadIdx.x & 31;
  const long long row = (long long)blockIdx.x * 8 + wave;
  const float* h = emb + row * DD;
  float s = 0.f;
#pragma unroll 4
  for (int i = lane; i < DD; i += 32) { float x = h[i]; s += x * x; }
#pragma unroll
  for (int off = 16; off > 0; off >>= 1) s += __shfl_xor(s, off, 32);
  if (lane == 0) rnorm[row] = 1.f / fmaxf(sqrtf(s), 1e-8f);
}

// ---------------------------------------------------------------------------
// LDS-staged, double-buffered fp32 WMMA GEMM.
//   C[M,N] = A[M,K] * op(B)   (BT: op(B)=B^T, B row-major [N,K];
//                              !BT: op(B)=B,  B row-major [K,N])
// EPI: 0 plain | 1 cosine+threshold+diag | 2 bias+relu
// Block = 128 threads = 4 waves. Block tile 32(M) x 256(N), K-chunk 16.
// Wave tile 32 x 64 -> 8 WMMA accumulators.
// Grid: (N/256, M/32, NWIN); all divide exactly -> EXEC all-ones for WMMA.
// ---------------------------------------------------------------------------
template <int EPI, bool BT>
__global__ __launch_bounds__(128) void gemm16_kernel(
    const float* __restrict__ Aall, long long aStrideZ,
    const float* __restrict__ Ball, long long bStrideZ,
    float* __restrict__ Call, long long cStrideZ,
    int lda, int ldb, int ldc, int K,
    const float* __restrict__ auxAll, long long auxStrideZ) {
  constexpr int KC = 16;                       // K-chunk
  constexpr int SA = 20;                       // lds stride [m][k] (pad: 16+4)
  constexpr int SB = BT ? 20 : 260;            // lds stride: [n][k] or [k][n]
  constexpr int ASZ = 32 * SA;                 // 640 floats
  constexpr int BSZ = BT ? 256 * 20 : 16 * 260;
  __shared__ float lsA[2][ASZ];
  __shared__ float lsB[2][BSZ];

  const int z = blockIdx.z;
  const float* A = Aall + (long long)z * aStrideZ;
  const float* B = Ball + (long long)z * bStrideZ;
  float* C = Call + (long long)z * cStrideZ;
  const float* aux = auxAll + (long long)z * auxStrideZ;

  const int tid  = threadIdx.x;
  const int wave = tid >> 5;
  const int lane = tid & 31;
  const int half = lane >> 4;                  // 0: lanes 0-15, 1: lanes 16-31
  const int lm   = lane & 15;
  const int m0blk = blockIdx.y * 32;
  const int n0blk = blockIdx.x * 256;
  const int nw    = wave * 64;                 // wave's N offset inside block

  // Issue async global->LDS copies for one K-chunk (fire and forget).
  auto prefetch = [&](int k0, int buf) {
    {  // A panel: 32 x 16 floats = 128 x b128
      const int row = tid >> 2, c4 = (tid & 3) << 2;
      async_cp16((unsigned)(size_t)&lsA[buf][row * SA + c4],
                 A + (long long)(m0blk + row) * lda + k0 + c4);
    }
    if (BT) {  // B panel [n][k]: 256 x 16 floats = 1024 x b128
#pragma unroll
      for (int i = 0; i < 8; ++i) {
        const int idx = tid + i * 128;
        const int row = idx >> 2, c4 = (idx & 3) << 2;
        async_cp16((unsigned)(size_t)&lsB[buf][row * 20 + c4],
                   B + (long long)(n0blk + row) * ldb + k0 + c4);
      }
    } else {   // B panel [k][n]: 16 x 256 floats = 1024 x b128
#pragma unroll
      for (int i = 0; i < 8; ++i) {
        const int idx = tid + i * 128;
        const int row = idx >> 6, c4 = (idx & 63) << 2;
        async_cp16((unsigned)(size_t)&lsB[buf][row * 260 + c4],
                   B + (long long)(k0 + row) * ldb + n0blk + c4);
      }
    }
  };

  v8f acc[2][4];
#pragma unroll
  for (int s = 0; s < 2; ++s)
#pragma unroll
    for (int j = 0; j < 4; ++j) acc[s][j] = v8f{};

  prefetch(0, 0);
  async_wait0();
  __syncthreads();

  int cur = 0;
  for (int k0 = 0; k0 < K; k0 += KC) {
    const bool hasNext = (k0 + KC) < K;
    if (hasNext) prefetch(k0 + KC, cur ^ 1);   // overlaps with WMMAs below

#pragma unroll
    for (int k = 0; k < KC; k += 4) {
      const int kk = k + 2 * half;             // lane lm holds K = kk, kk+1
      v2f a0 = *(const v2f*)&lsA[cur][(0  + lm) * SA + kk];
      v2f a1 = *(const v2f*)&lsA[cur][(16 + lm) * SA + kk];
      v2f b[4];
#pragma unroll
      for (int j = 0; j < 4; ++j) {
        if (BT) {
          b[j] = *(const v2f*)&lsB[cur][(nw + j * 16 + lm) * 20 + kk];
        } else {
          b[j].x = lsB[cur][(kk)     * 260 + nw + j * 16 + lm];
          b[j].y = lsB[cur][(kk + 1) * 260 + nw + j * 16 + lm];
        }
      }
#pragma unroll
      for (int j = 0; j < 4; ++j) {
        acc[0][j] = __builtin_amdgcn_wmma_f32_16x16x4_f32(false, a0, false, b[j], (short)0, acc[0][j], false, false);
        acc[1][j] = __builtin_amdgcn_wmma_f32_16x16x4_f32(false, a1, false, b[j], (short)0, acc[1][j], false, false);
      }
    }

    if (hasNext) {
      async_wait0();       // this wave's async copies have landed in LDS
      __syncthreads();     // ds reads above already consumed (reg deps)
      cur ^= 1;
    }
  }

  // Epilogue + store. C layout: VGPR r -> row = r + 8*half, lanes -> col.
#pragma unroll
  for (int s = 0; s < 2; ++s) {
#pragma unroll
    for (int j = 0; j < 4; ++j) {
      const int col = n0blk + nw + j * 16 + lm;
      float auxc = 0.f;
      if (EPI == 1 || EPI == 2) auxc = aux[col];
#pragma unroll
      for (int r = 0; r < 8; ++r) {
        const int row = m0blk + s * 16 + r + 8 * half;
        float v = acc[s][j][r];
        if (EPI == 1) {
          v *= aux[row] * auxc;                // cosine similarity
          v = (v > 0.3f) ? v : 0.f;            // threshold
          if (row == col) v = 1.f;             // eye diagonal
        } else if (EPI == 2) {
          v += auxc;                           // bias
          v = (v > 0.f) ? v : 0.f;             // relu
        }
        C[(long long)row * ldc + col] = v;
      }
    }
  }
}

// ---------------------------------------------------------------------------
// win_emb[w][d] = mean_t h2[w][t][d]
// ---------------------------------------------------------------------------
__global__ __launch_bounds__(256) void colmean_kernel(const float* __restrict__ h2,
                                                      float* __restrict__ we) {
  const int d = blockIdx.x * 256 + threadIdx.x;
  const int w = blockIdx.y;
  const float* p = h2 + (long long)w * TT * DD + d;
  float s = 0.f;
#pragma unroll 4
  for (int t = 0; t < TT; ++t) s += p[(long long)t * DD];
  we[(long long)w * DD + d] = s * (1.f / TT);
}

// wbar[b][d] = mean_w we[b*16+w][d]  (mean commutes with the final Linear)
__global__ __launch_bounds__(256) void wbar_kernel(const float* __restrict__ we,
                                                   float* __restrict__ wbar) {
  const int d = blockIdx.x * 256 + threadIdx.x;
  const int b = blockIdx.y;
  float s = 0.f;
#pragma unroll
  for (int w = 0; w < 16; ++w) s += we[(long long)(b * 16 + w) * DD + d];
  wbar[(long long)b * DD + d] = s * (1.f / 16.f);
}

// out[b][d] = wbar[b] . Wg[d] + bg[d]
__global__ __launch_bounds__(256) void final_kernel(const float* __restrict__ wbar,
                                                    const float* __restrict__ Wg,
                                                    const float* __restrict__ bg,
                                                    float* __restrict__ out) {
  const int d = blockIdx.x * 256 + threadIdx.x;
  const int b = blockIdx.y;
  const float* wg = Wg + (long long)d * DD;
  const float* wb = wbar + (long long)b * DD;
  float acc = 0.f;
#pragma unroll 4
  for (int k = 0; k < DD; ++k) acc += wb[k] * wg[k];
  out[(long long)b * DD + d] = acc + bg[d];
}

// ---------------------------------------------------------------------------
extern "C" void kernel_launch(void* const* d_in, const int* in_sizes, int n_in,
                              void* d_out, int out_size, void* d_ws, size_t ws_size,
                              hipStream_t stream) {
  (void)in_sizes; (void)n_in; (void)out_size; (void)ws_size;
  const float* emb = (const float*)d_in[0];
  const float* W1  = (const float*)d_in[1];
  const float* b1  = (const float*)d_in[2];
  const float* W2  = (const float*)d_in[3];
  const float* b2  = (const float*)d_in[4];
  const float* Wg  = (const float*)d_in[5];
  const float* bg  = (const float*)d_in[6];
  float* out = (float*)d_out;

  // Workspace layout (floats): rnorm | adj | bufA | bufB | we | wbar (~256 MB)
  float* ws = (float*)d_ws;
  const long long szRN  = (long long)NWIN * TT;       // 32768
  const long long szADJ = (long long)NWIN * TT * TT;  // 16 Mi
  const long long szBUF = (long long)NWIN * TT * DD;  // 24 Mi
  float* rnorm = ws;
  float* adj   = rnorm + szRN;
  float* bufA  = adj + szADJ;
  float* bufB  = bufA + szBUF;
  float* we    = bufB + szBUF;
  float* wbar  = we + (long long)NWIN * DD;

  // 1: reciprocal row norms
  rnorm_kernel<<<dim3((NWIN * TT) / 8), 256, 0, stream>>>(emb, rnorm);

  // 2: adj = diag1(threshold((h h^T) * rn_i * rn_j))      [512x512, K=768]
  gemm16_kernel<1, true><<<dim3(TT / 256, TT / 32, NWIN), 128, 0, stream>>>(
      emb, (long long)TT * DD, emb, (long long)TT * DD, adj, (long long)TT * TT,
      DD, DD, TT, DD, rnorm, TT);

  // 3: t1 = adj @ h                                        [512x768, K=512]
  gemm16_kernel<0, false><<<dim3(DD / 256, TT / 32, NWIN), 128, 0, stream>>>(
      adj, (long long)TT * TT, emb, (long long)TT * DD, bufA, (long long)TT * DD,
      TT, DD, DD, TT, adj /*unused*/, 0);

  // 4: h1 = relu(t1 @ W1^T + b1)                           [512x768, K=768]
  gemm16_kernel<2, true><<<dim3(DD / 256, TT / 32, NWIN), 128, 0, stream>>>(
      bufA, (long long)TT * DD, W1, 0, bufB, (long long)TT * DD,
      DD, DD, DD, DD, b1, 0);

  // 5: t2 = adj @ h1
  gemm16_kernel<0, false><<<dim3(DD / 256, TT / 32, NWIN), 128, 0, stream>>>(
      adj, (long long)TT * TT, bufB, (long long)TT * DD, bufA, (long long)TT * DD,
      TT, DD, DD, TT, adj /*unused*/, 0);

  // 6: h2 = relu(t2 @ W2^T + b2)
  gemm16_kernel<2, true><<<dim3(DD / 256, TT / 32, NWIN), 128, 0, stream>>>(
      bufA, (long long)TT * DD, W2, 0, bufB, (long long)TT * DD,
      DD, DD, DD, DD, b2, 0);

  // 7: per-window token mean
  colmean_kernel<<<dim3(DD / 256, NWIN), 256, 0, stream>>>(bufB, we);

  // 8: mean over the 16 windows of each batch
  wbar_kernel<<<dim3(DD / 256, 4), 256, 0, stream>>>(we, wbar);

  // 9: out = wbar @ Wg^T + bg
  final_kernel<<<dim3(DD / 256, 4), 256, 0, stream>>>(wbar, Wg, bg, out);
}